// NTNDecoder_8710193677085
// MI455X (gfx1250) — compile-verified
//
#include <hip/hip_runtime.h>

// ---------------------------------------------------------------------------
// NTN decoder: logits[b,j,m] = U . tanh( x_j W x_m + Wl x_j + Wr x_m + Bvec )
// B=32, L=128, D=128, H=128.  f32 WMMA (16x16x4) path on gfx1250.
// ---------------------------------------------------------------------------

typedef float v2f __attribute__((ext_vector_type(2)));
typedef float v8f __attribute__((ext_vector_type(8)));

#define NB 32
#define NL 128
#define ND 128
#define NH 128
#define NEG_BIG 1.0e8f

static __device__ __forceinline__ v8f wmma_f32(v2f a, v2f b, v8f c) {
    return __builtin_amdgcn_wmma_f32_16x16x4_f32(false, a, false, b, (short)0, c,
                                                 false, false);
}

// --------------------------- prep kernels ----------------------------------

// Pack W into WMMA B-fragment order:
//   Wt[h][d2][e][q] = W[2*d2+q][e][h]
// so a lane's K-pair (W[d0,e,h], W[d0+1,e,h]) is one aligned 8-byte load and
// a half-wave covers a contiguous 128B segment.
__global__ void packW_kernel(const float* __restrict__ W,
                             float* __restrict__ Wt) {
    int i = blockIdx.x * 256 + threadIdx.x;  // output: h*16384 + d2*256 + e*2 + q
    if (i >= ND * ND * NH) return;
    int q  = i & 1;
    int e  = (i >> 1) & 127;
    int d2 = (i >> 8) & 63;
    int h  = i >> 14;
    int d  = (d2 << 1) | q;
    Wt[i] = W[(d << 14) + (e << 7) + h];
}

// dL[row][h] = sum_d X[row][d] * Wl[d][h];  dR likewise with Wr. row in [0,B*L)
__global__ void dots_kernel(const float* __restrict__ X,
                            const float* __restrict__ Wl,
                            const float* __restrict__ Wr,
                            float* __restrict__ dL,
                            float* __restrict__ dR) {
    int i = blockIdx.x * 256 + threadIdx.x;   // sel(1) | row(12) | h(7)
    int h   = i & 127;
    int row = (i >> 7) & 4095;
    int sel = i >> 19;
    const float* Wm = sel ? Wr : Wl;
    const float* xr = X + ((size_t)row << 7);
    float s = 0.f;
#pragma unroll 8
    for (int d = 0; d < ND; ++d) s = fmaf(xr[d], Wm[(d << 7) + h], s);
    (sel ? dR : dL)[(row << 7) + h] = s;
}

// --------------------------- RNG (JAX threefry2x32, key(42)) ---------------

static __device__ __forceinline__ unsigned int rotl32(unsigned int x, int r) {
    return (x << r) | (x >> (32 - r));
}

// Reproduces jax.random.uniform bits for flat index fi over 2^19 elements
// with key = (0, 42): counts split into halves x0 = i, x1 = i + half.
static __device__ unsigned int threefry_bits(unsigned int fi) {
    const unsigned int half = (NB * NL * NL) / 2;  // 262144
    unsigned int x0, x1;
    bool first;
    if (fi < half) { x0 = fi;        x1 = fi + half; first = true;  }
    else           { x0 = fi - half; x1 = fi;        first = false; }
    const unsigned int k0 = 0u, k1 = 42u;
    const unsigned int k2 = k0 ^ k1 ^ 0x1BD11BDAu;
    const int RA[4] = {13, 15, 26, 6};
    const int RB[4] = {17, 29, 16, 24};
    x0 += k0; x1 += k1;
#pragma unroll
    for (int i = 0; i < 4; ++i) { x0 += x1; x1 = rotl32(x1, RA[i]); x1 ^= x0; }
    x0 += k1; x1 += k2 + 1u;
#pragma unroll
    for (int i = 0; i < 4; ++i) { x0 += x1; x1 = rotl32(x1, RB[i]); x1 ^= x0; }
    x0 += k2; x1 += k0 + 2u;
#pragma unroll
    for (int i = 0; i < 4; ++i) { x0 += x1; x1 = rotl32(x1, RA[i]); x1 ^= x0; }
    x0 += k0; x1 += k1 + 3u;
#pragma unroll
    for (int i = 0; i < 4; ++i) { x0 += x1; x1 = rotl32(x1, RB[i]); x1 ^= x0; }
    x0 += k1; x1 += k2 + 4u;
#pragma unroll
    for (int i = 0; i < 4; ++i) { x0 += x1; x1 = rotl32(x1, RA[i]); x1 ^= x0; }
    x0 += k2; x1 += k0 + 5u;
    return first ? x0 : x1;
}

// --------------------------- main fused kernel -----------------------------
// grid = (L/16, B), block = 256 (8 waves). Wave w handles h in [16w, 16w+16).
__launch_bounds__(256, 1)
__global__ void ntn_main_kernel(const float* __restrict__ X,   // [B,L,D]
                                const float* __restrict__ Wt,  // packed W
                                const float* __restrict__ dL,  // [B*L,H]
                                const float* __restrict__ dR,  // [B*L,H]
                                const float* __restrict__ U,   // [H]
                                const float* __restrict__ Bvec,// [H]
                                const float* __restrict__ lbias,
                                float* __restrict__ out)       // 3x[B,L,L]
{
    __shared__ float Xt[ND * 129];        // Xt[e*129+m] = X[b][m][e] (padded)
    __shared__ float Tbuf[8][16 * 130];   // per-wave T[j][e] staging (padded)
    __shared__ float Lg[16 * NL];         // logits accumulator for this j-tile

    const int b    = blockIdx.y;
    const int j0   = blockIdx.x << 4;
    const int t    = threadIdx.x;
    const int wave = t >> 5;
    const int lane = t & 31;
    const int lh   = lane & 15;
    const bool lo  = lane < 16;

    const float* Xb = X + ((size_t)b << 14);

    // Load X_b transposed into LDS (coalesced global reads), zero Lg.
    for (int idx = t; idx < NL * ND; idx += 256) {
        int m = idx >> 7, e = idx & 127;
        Xt[e * 129 + m] = Xb[idx];
    }
    for (int idx = t; idx < 16 * NL; idx += 256) Lg[idx] = 0.f;
    __syncthreads();

    const v8f vzero = {0.f, 0.f, 0.f, 0.f, 0.f, 0.f, 0.f, 0.f};
    v8f acc[8];
#pragma unroll
    for (int n = 0; n < 8; ++n) acc[n] = vzero;

    float* Tw = &Tbuf[wave][0];

    for (int hi = 0; hi < 16; ++hi) {
        const int h = (wave << 4) + hi;
        const float* Wh = Wt + ((size_t)h << 14);

        // Hint-prefetch the next W_h slab into cache.
        if (hi < 15) {
            const float* Wn = Wh + (1 << 14);
            __builtin_prefetch(Wn + (lane << 4), 0, 3);
            __builtin_prefetch(Wn + 8192 + (lane << 4), 0, 3);
        }

        // ---- Stage A: T(16x128) = X[j0:j0+16, :] @ W_h ----
        v8f T[8];
#pragma unroll
        for (int n = 0; n < 8; ++n) T[n] = vzero;
#pragma unroll 2
        for (int kk = 0; kk < 32; ++kk) {
            const int d0 = (kk << 2) + (lo ? 0 : 2);   // K-pair per half-wave
            const int d2 = d0 >> 1;
            v2f a;
            a.x = Xt[d0 * 129 + j0 + lh];
            a.y = Xt[(d0 + 1) * 129 + j0 + lh];
            const float* wrow = Wh + (d2 << 8);        // 256 floats per d2 row
#pragma unroll
            for (int n = 0; n < 8; ++n) {
                // one aligned b64: (W[d0, e, h], W[d0+1, e, h])
                const v2f bb = *(const v2f*)(wrow + ((((n << 4) + lh)) << 1));
                T[n] = wmma_f32(a, bb, T[n]);
            }
        }
        // Spill T (C/D layout) to LDS as row-major T[j][e] for A-fragments.
#pragma unroll
        for (int n = 0; n < 8; ++n) {
            const int e = (n << 4) + lh;
#pragma unroll
            for (int r = 0; r < 8; ++r) {
                const int j = lo ? r : (r + 8);
                Tw[j * 130 + e] = T[n][r];
            }
        }
        __syncthreads();

        // ---- Stage B: S(16x128) = T @ X_b^T ----
        v8f S[8];
#pragma unroll
        for (int n = 0; n < 8; ++n) S[n] = vzero;
#pragma unroll 2
        for (int kk = 0; kk < 32; ++kk) {
            const int e0 = (kk << 2) + (lo ? 0 : 2);
            v2f a;
            a.x = Tw[lh * 130 + e0];
            a.y = Tw[lh * 130 + e0 + 1];
#pragma unroll
            for (int n = 0; n < 8; ++n) {
                v2f bb;
                bb.x = Xt[e0 * 129 + (n << 4) + lh];
                bb.y = Xt[(e0 + 1) * 129 + (n << 4) + lh];
                S[n] = wmma_f32(a, bb, S[n]);
            }
        }

        // ---- fuse: acc += U[h] * tanh(S + dot_l[j,h] + dot_r[m,h] + Bvec[h])
        const float Uh = U[h];
        const float Bh = Bvec[h];
        float dl[8];
#pragma unroll
        for (int r = 0; r < 8; ++r) {
            const int j = j0 + (lo ? r : (r + 8));
            dl[r] = dL[((((size_t)b << 7) + j) << 7) + h];
        }
#pragma unroll
        for (int n = 0; n < 8; ++n) {
            const int m = (n << 4) + lh;
            const float dr = dR[((((size_t)b << 7) + m) << 7) + h];
#pragma unroll
            for (int r = 0; r < 8; ++r) {
                const float z = S[n][r] + dl[r] + dr + Bh;
                acc[n][r] += Uh * tanhf(z);
            }
        }
        __syncthreads();   // Tw reused next iteration
    }

    // Reduce the 8 waves' partial h-sums into Lg.
#pragma unroll
    for (int n = 0; n < 8; ++n) {
        const int m = (n << 4) + lh;
#pragma unroll
        for (int r = 0; r < 8; ++r) {
            const int j = lo ? r : (r + 8);
            atomicAdd(&Lg[j * NL + m], acc[n][r]);
        }
    }
    __syncthreads();

    // Finalize: mask, sigmoid, entropy, Bernoulli samples.
    const float lb = lbias[0];
    for (int idx = t; idx < 16 * NL; idx += 256) {
        const int j  = idx >> 7;
        const int m  = idx & 127;
        const int jg = j0 + j;
        float z = Lg[idx] + lb;
        if (jg == m) z -= NEG_BIG;
        const float p  = 1.0f / (1.0f + expf(-z));
        const float sp = fmaxf(z, 0.f) + log1pf(expf(-fabsf(z)));
        const float ent = sp - z * p;

        const unsigned int fi =
            ((unsigned int)b << 14) | ((unsigned int)jg << 7) | (unsigned int)m;
        const unsigned int bits = threefry_bits(fi);
        const float u = __uint_as_float((bits >> 9) | 0x3f800000u) - 1.0f;
        const float sample = (u < p) ? 1.0f : 0.0f;

        out[fi] = sample;                        // samples
        out[(NB * NL * NL) + fi] = z;            // mask_scores
        out[2 * (NB * NL * NL) + fi] = ent;      // entropy
    }
}

// --------------------------- launcher --------------------------------------

extern "C" void kernel_launch(void* const* d_in, const int* in_sizes, int n_in,
                              void* d_out, int out_size, void* d_ws,
                              size_t ws_size, hipStream_t stream) {
    (void)in_sizes; (void)n_in; (void)out_size; (void)ws_size;
    const float* X   = (const float*)d_in[0];   // [B,L,D]
    const float* W   = (const float*)d_in[1];   // [D,D,H]
    const float* Wl  = (const float*)d_in[2];   // [D,H]
    const float* Wr  = (const float*)d_in[3];   // [D,H]
    const float* U   = (const float*)d_in[4];   // [H]
    const float* Bv  = (const float*)d_in[5];   // [H]
    const float* lb  = (const float*)d_in[6];   // [1]
    float* out = (float*)d_out;

    float* ws = (float*)d_ws;
    float* Wt = ws;                               // 2,097,152 f32 (8 MB)
    float* dL = ws + 2097152;                     //   524,288 f32 (2 MB)
    float* dR = ws + 2097152 + 524288;            //   524,288 f32 (2 MB)

    packW_kernel<<<(ND * ND * NH) / 256, 256, 0, stream>>>(W, Wt);
    dots_kernel<<<(2 * NB * NL * NH) / 256, 256, 0, stream>>>(X, Wl, Wr, dL, dR);

    dim3 grid(NL / 16, NB);
    ntn_main_kernel<<<grid, 256, 0, stream>>>(X, Wt, dL, dR, U, Bv, lb, out);
}